// IMUHOIModel_48052094107762
// MI455X (gfx1250) — compile-verified
//
#include <hip/hip_runtime.h>
#include <math.h>

// ---------------------------------------------------------------------------
// fk_obj_trans on MI455X (gfx1250).
//
// Roofline: 232 MB streamed, ~126 MFLOP -> pure bandwidth/latency problem
// (~10us floor at 23.3 TB/s). One wave32 per batch row (4096 waves); the
// wave acts as scalar processor (SALU recurrence) + DMA engine:
// 32-step chunks staged global->LDS with global_load_async_to_lds_b128,
// 3-deep pipeline (2 chunks in flight) synchronized by s_wait_asynccnt,
// coalesced 96-float chunk stores.
// ---------------------------------------------------------------------------

#define FK_S      512
#define FK_CH     32                 // timesteps per chunk
#define FK_NBUF   3                  // pipeline depth (2 chunks in flight)
#define FK_THRESH 0.5f
#define FK_DEFLEN 0.1f
#define FK_EPS    1e-6f

// float4 slots per chunk: prob 64f=16, hands 192f=48, rotm 288f=72
#define FK_PROB_F4  16
#define FK_HAND_F4  48
#define FK_ROTM_F4  72
#define FK_HAND_OFF 16               // FK_PROB_F4
#define FK_ROTM_OFF 64               // FK_PROB_F4 + FK_HAND_F4
#define FK_CHUNK_F4 136

#if __has_builtin(__builtin_amdgcn_global_load_async_to_lds_b128) && \
    __has_builtin(__builtin_amdgcn_s_wait_asynccnt)
#define FK_ASYNC 1
#else
#define FK_ASYNC 0
#endif

typedef int fk_v4i __attribute__((ext_vector_type(4)));
typedef __attribute__((address_space(1))) fk_v4i* fk_g4p;   // global int4*
typedef __attribute__((address_space(3))) fk_v4i* fk_l4p;   // LDS int4*

__device__ __forceinline__ void fk_cp16(float4* l, const float4* g) {
#if FK_ASYNC
  // (global src, lds dst, imm offset, imm cpol)
  __builtin_amdgcn_global_load_async_to_lds_b128((fk_g4p)g, (fk_l4p)l, 0, 0);
#else
  *l = *g;
#endif
}

// Exactly 6 async instructions per chunk (wave-level ASYNCcnt += 6).
__device__ __forceinline__ void fk_issue_chunk(float4* buf,
                                               const float4* pp,
                                               const float4* hp,
                                               const float4* rp,
                                               int c, int lane) {
  pp += (size_t)c * FK_PROB_F4;
  hp += (size_t)c * FK_HAND_F4;
  rp += (size_t)c * FK_ROTM_F4;
  if (lane < FK_PROB_F4)        fk_cp16(buf + lane,                    pp + lane);
  fk_cp16(buf + FK_HAND_OFF + lane,      hp + lane);
  if (lane < FK_HAND_F4 - 32)   fk_cp16(buf + FK_HAND_OFF + 32 + lane, hp + 32 + lane);
  fk_cp16(buf + FK_ROTM_OFF + lane,      rp + lane);
  fk_cp16(buf + FK_ROTM_OFF + 32 + lane, rp + 32 + lane);
  if (lane < FK_ROTM_F4 - 64)   fk_cp16(buf + FK_ROTM_OFF + 64 + lane, rp + 64 + lane);
}

__global__ __launch_bounds__(32)
void fk_obj_trans_kernel(const float* __restrict__ prob,
                         const float* __restrict__ hands,
                         const float* __restrict__ rotm,
                         const float* __restrict__ tinit,
                         float* __restrict__ out,
                         int B) {
  __shared__ float4 buf[FK_NBUF][FK_CHUNK_F4];
  __shared__ float  obuf[FK_CH * 3];

  const int b = (int)blockIdx.x;
  if (b >= B) return;
  const int lane = (int)threadIdx.x;

  const size_t rowbase = (size_t)b * FK_S;
  const float4* pp = (const float4*)(prob  + rowbase * 2);
  const float4* hp = (const float4*)(hands + rowbase * 6);
  const float4* rp = (const float4*)(rotm  + rowbase * 9);
  float* orow = out + rowbase * 3;

  // carried state (wave-uniform -> SALU; all lanes redundant)
  int   cur = -1;
  float px = tinit[b * 3 + 0];
  float py = tinit[b * 3 + 1];
  float pz = tinit[b * 3 + 2];
  float dlx = 0.f, dly = 0.f, dlz = 0.f, dist = 0.f;
  bool  lcp = false, rcp = false;

  const int NCH = FK_S / FK_CH;     // 16

  // prime the pipeline: chunks 0 and 1 in flight
  fk_issue_chunk(buf[0], pp, hp, rp, 0, lane);
  fk_issue_chunk(buf[1], pp, hp, rp, 1, lane);
#if !FK_ASYNC
  __syncthreads();
#endif

  for (int c = 0; c < NCH; ++c) {
    const int cb = c % FK_NBUF;

    __syncthreads();  // prior chunk's LDS reads + obuf traffic done (single-wave: waits + S_NOP)

    if (c + 2 < NCH) {
      // refill the buffer consumed in the previous iteration
      fk_issue_chunk(buf[(c + 2) % FK_NBUF], pp, hp, rp, c + 2, lane);
#if FK_ASYNC
      __builtin_amdgcn_s_wait_asynccnt(12);  // chunk c landed; c+1, c+2 in flight
#else
      __syncthreads();
#endif
    } else if (c + 1 < NCH) {
#if FK_ASYNC
      __builtin_amdgcn_s_wait_asynccnt(6);   // chunk c landed; c+1 in flight
#endif
    } else {
#if FK_ASYNC
      __builtin_amdgcn_s_wait_asynccnt(0);   // last chunk landed
#endif
    }

    const float* P = (const float*)(buf[cb]);                 // 64 floats
    const float* H = (const float*)(buf[cb] + FK_HAND_OFF);   // 192 floats
    const float* R = (const float*)(buf[cb] + FK_ROTM_OFF);   // 288 floats
    const int s0 = c * FK_CH;

    for (int t = 0; t < FK_CH; ++t) {
      const int s = s0 + t;
      const bool first = (s == 0);
      const bool lc = (P[2 * t]     > FK_THRESH) && !first;
      const bool rc = (P[2 * t + 1] > FK_THRESH) && !first;
      const bool ls = lc && !lcp;
      const bool rs = rc && !rcp;
      lcp = lc; rcp = rc;

      const int  nw     = ls ? 0 : (rs ? 1 : -1);
      const bool has_c  = (cur == 0) ? lc : ((cur == 1) ? rc : false);
      const bool has_o  = (cur == 0) ? rc : ((cur == 1) ? lc : false);
      const bool regrab = (nw != -1);
      const bool sw     = !regrab && !has_c && has_o;
      const bool rel    = !regrab && !sw && !has_c && (cur != -1);
      const int  ncur   = regrab ? nw : (sw ? (1 - cur) : (rel ? -1 : cur));
      const int  hidx   = (ncur < 0) ? 0 : ((ncur > 1) ? 1 : ncur);

      const float* h0 = H + t * 6 + hidx * 3;
      const float hx = h0[0], hy = h0[1], hz = h0[2];

      const float vx = px - hx, vy = py - hy, vz = pz - hz;
      const float d0 = sqrtf(vx * vx + vy * vy + vz * vz);
      const bool  small = (d0 <= FK_EPS);
      const float inv = small ? 0.f : (1.0f / d0);
      const float ux = small ? 0.f : vx * inv;
      const float uy = small ? 0.f : vy * inv;
      const float uz = small ? 1.f : vz * inv;
      const float dd = small ? FK_DEFLEN : d0;

      const float* Rt = R + t * 9;
      const float r00 = Rt[0], r01 = Rt[1], r02 = Rt[2];
      const float r10 = Rt[3], r11 = Rt[4], r12 = Rt[5];
      const float r20 = Rt[6], r21 = Rt[7], r22 = Rt[8];

      // cand = R^T * unit
      const float cx = r00 * ux + r10 * uy + r20 * uz;
      const float cy = r01 * ux + r11 * uy + r21 * uz;
      const float cz = r02 * ux + r12 * uy + r22 * uz;

      const bool rec = regrab || sw;
      dlx  = rec ? cx : dlx;
      dly  = rec ? cy : dly;
      dlz  = rec ? cz : dlz;
      dist = rec ? dd : dist;

      // pos = attached ? hand + (R * dloc) * dist : pos
      const bool att = (ncur != -1);
      const float wx = r00 * dlx + r01 * dly + r02 * dlz;
      const float wy = r10 * dlx + r11 * dly + r12 * dlz;
      const float wz = r20 * dlx + r21 * dly + r22 * dlz;
      px = att ? (hx + wx * dist) : px;
      py = att ? (hy + wy * dist) : py;
      pz = att ? (hz + wz * dist) : pz;
      cur = ncur;

      if (lane == 0) {
        obuf[3 * t + 0] = px;
        obuf[3 * t + 1] = py;
        obuf[3 * t + 2] = pz;
      }
    }

    __syncthreads();
    // coalesced store of 96 floats (32 lanes x 3)
    float* od = orow + (size_t)s0 * 3;
    od[lane]      = obuf[lane];
    od[lane + 32] = obuf[lane + 32];
    od[lane + 64] = obuf[lane + 64];
  }
}

extern "C" void kernel_launch(void* const* d_in, const int* in_sizes, int n_in,
                              void* d_out, int out_size, void* d_ws, size_t ws_size,
                              hipStream_t stream) {
  (void)n_in; (void)out_size; (void)d_ws; (void)ws_size;
  const float* prob  = (const float*)d_in[0];  // [B,S,2]
  const float* hands = (const float*)d_in[1];  // [B,S,2,3]
  const float* rotm  = (const float*)d_in[2];  // [B,S,3,3]
  const float* tinit = (const float*)d_in[3];  // [B,3]
  float* out = (float*)d_out;                  // [B,S,3]

  const int B = in_sizes[3] / 3;               // 4096
  fk_obj_trans_kernel<<<dim3((unsigned)B), dim3(32), 0, stream>>>(
      prob, hands, rotm, tinit, out, B);
}